// UnitedNormBase_89558658056883
// MI455X (gfx1250) — compile-verified
//
#include <hip/hip_runtime.h>
#include <cstdint>

typedef __attribute__((ext_vector_type(2))) float v2f;
typedef __attribute__((ext_vector_type(4))) float v4f;
typedef __attribute__((ext_vector_type(8))) float v8f;

#define NN 100000
#define FF 128
#define EE 600000
#define GG 1000
#define EPSF 1e-5f

// ---- workspace layout (in floats) ----
static constexpr size_t OFF_BSUM = 0;                        // [FF]
static constexpr size_t OFF_BSQ  = OFF_BSUM + FF;            // [FF]
static constexpr size_t OFF_GCNT = OFF_BSQ + FF;             // [GG]
static constexpr size_t OFF_GSUM = OFF_GCNT + GG;            // [GG*FF]
static constexpr size_t OFF_GSQ  = OFF_GSUM + (size_t)GG*FF; // [GG*FF]
static constexpr size_t OFF_DEG  = OFF_GSQ  + (size_t)GG*FF; // [NN]
static constexpr size_t OFF_AS   = OFF_DEG  + NN;            // [NN*FF]
static constexpr size_t OFF_AS2  = OFF_AS   + (size_t)NN*FF; // [NN*FF]
static constexpr size_t WS_FLOATS = OFF_AS2 + (size_t)NN*FF;

__device__ __forceinline__ v4f max4(v4f a, v4f b) {
  return (v4f){fmaxf(a.x,b.x), fmaxf(a.y,b.y), fmaxf(a.z,b.z), fmaxf(a.w,b.w)};
}
__device__ __forceinline__ v4f rsqrt4(v4f a) {
  return (v4f){rsqrtf(a.x), rsqrtf(a.y), rsqrtf(a.z), rsqrtf(a.w)};
}
__device__ __forceinline__ v4f exp4(v4f a) {
  return (v4f){__expf(a.x), __expf(a.y), __expf(a.z), __expf(a.w)};
}

// ---------------------------------------------------------------------------
// K1: batch-norm sums + per-graph segment sums. graph_id is sorted, so each
// thread run-length accumulates per graph and flushes atomics only on segment
// boundaries (~(G + #blocks) * 256 atomics total).
// Block = 256 threads = 128 features x 2 row-slots; block covers 64 rows.
// ---------------------------------------------------------------------------
__global__ __launch_bounds__(256) void k_stats(const float* __restrict__ x,
    const int* __restrict__ gidv,
    float* __restrict__ bsum, float* __restrict__ bsq,
    float* __restrict__ gcnt, float* __restrict__ gsum, float* __restrict__ gsq) {
  const int tf = threadIdx.x & 127;
  const int tr = threadIdx.x >> 7;           // 0..1
  const int base = blockIdx.x * 64;
  float bs = 0.0f, bs2 = 0.0f;
  int curg = -1;
  float gs = 0.0f, gs2 = 0.0f, gc = 0.0f;
  for (int r = tr; r < 64; r += 2) {
    const int row = base + r;
    if (row >= NN) break;
    const int gid = gidv[row];
    if (gid != curg) {
      if (curg >= 0) {
        atomicAdd(&gsum[(size_t)curg * FF + tf], gs);
        atomicAdd(&gsq [(size_t)curg * FF + tf], gs2);
        if (tf == 0) atomicAdd(&gcnt[curg], gc);
      }
      curg = gid; gs = 0.0f; gs2 = 0.0f; gc = 0.0f;
    }
    const float v = x[(size_t)row * FF + tf];
    bs += v; bs2 += v * v;
    gs += v; gs2 += v * v; gc += 1.0f;
  }
  if (curg >= 0) {
    atomicAdd(&gsum[(size_t)curg * FF + tf], gs);
    atomicAdd(&gsq [(size_t)curg * FF + tf], gs2);
    if (tf == 0) atomicAdd(&gcnt[curg], gc);
  }
  atomicAdd(&bsum[tf], bs);
  atomicAdd(&bsq [tf], bs2);
}

// ---------------------------------------------------------------------------
// K2: edge gather/scatter. One wave per 4 edges; the 512B source row is staged
// into LDS with gfx1250 async-to-LDS loads (16B per lane, ASYNCcnt tracked),
// then scattered to per-dst accumulators with f32 global atomics.
// ---------------------------------------------------------------------------
__global__ __launch_bounds__(256) void k_edges(const float* __restrict__ x,
    const int* __restrict__ esrc, const int* __restrict__ edst,
    float* __restrict__ asum, float* __restrict__ asq, float* __restrict__ deg) {
  __shared__ __align__(16) float smem[8 * FF];   // 8 waves * one row each
  const int lane = threadIdx.x & 31;
  const int wv   = threadIdx.x >> 5;
  const int gw   = blockIdx.x * 8 + wv;          // global wave id
  float* mylds = smem + wv * FF + lane * 4;
  const uint32_t ldsoff = (uint32_t)(uintptr_t)mylds;  // low 32b of LDS generic addr
  for (int i = 0; i < 4; ++i) {
    const int e = gw * 4 + i;
    if (e >= EE) return;                         // wave-uniform exit
    const int src = esrc[e];
    const int dst = edst[e];
    const uint64_t gaddr = (uint64_t)(uintptr_t)(x + (size_t)src * FF + lane * 4);
    asm volatile("global_load_async_to_lds_b128 %0, %1, off"
                 :: "v"(ldsoff), "v"(gaddr) : "memory");
    asm volatile("s_wait_asynccnt 0" ::: "memory");
    const v4f v = *(const v4f*)mylds;
    float* s1 = asum + (size_t)dst * FF + lane * 4;
    float* s2 = asq  + (size_t)dst * FF + lane * 4;
    atomicAdd(s1 + 0, v.x);       atomicAdd(s1 + 1, v.y);
    atomicAdd(s1 + 2, v.z);       atomicAdd(s1 + 3, v.w);
    atomicAdd(s2 + 0, v.x * v.x); atomicAdd(s2 + 1, v.y * v.y);
    atomicAdd(s2 + 2, v.z * v.z); atomicAdd(s2 + 3, v.w * v.w);
    if (lane == 0) atomicAdd(deg + dst, 1.0f);
  }
}

// ---------------------------------------------------------------------------
// K3: final combine. Each wave owns a 16-row tile.
// Phase A: layer-norm row sums / sumsq via V_WMMA_F32_16X16X4_F32 with
//          B = ones(4x16): D[m][n] = sum_k A[m][k] (exact f32 row sums).
//          A layout (16x4 f32): lanes 0-15 -> K=k0,k0+1; lanes 16-31 -> K=k0+2,k0+3.
//          C/D layout (16x16 f32): lanes 0-15 VGPR j holds M=j; lanes 16-31 M=j+8.
// Phase B: per-lane float4 feature slice (f0=lane*4), fuse all 4 norms.
// ---------------------------------------------------------------------------
__global__ __launch_bounds__(256) void k_final(
    const float* __restrict__ x,
    const float* __restrict__ gamma, const float* __restrict__ beta,
    const float* __restrict__ lb, const float* __restrict__ lg,
    const float* __restrict__ la, const float* __restrict__ lnv,
    const int* __restrict__ gidv,
    const float* __restrict__ bsum, const float* __restrict__ bsq,
    const float* __restrict__ gcnt, const float* __restrict__ gsum,
    const float* __restrict__ gsq, const float* __restrict__ deg,
    const float* __restrict__ asum, const float* __restrict__ asq,
    float* __restrict__ out) {
  const int lane = threadIdx.x & 31;
  const int wv   = threadIdx.x >> 5;
  const int tile = blockIdx.x * 8 + wv;
  const int tb   = tile * 16;
  if (tb >= NN) return;                          // wave-uniform; EXEC stays all-1s

  // ---- Phase A: WMMA row reductions ----
  const int arow0 = tb + (lane & 15);
  const int arow  = arow0 < NN ? arow0 : NN - 1;
  const float zm  = arow0 < NN ? 1.0f : 0.0f;    // zero-pad partial tiles
  const float* xr = x + (size_t)arow * FF + ((lane >> 4) << 1);
  const v2f ones2 = {1.0f, 1.0f};
  v8f acc  = {0.f,0.f,0.f,0.f,0.f,0.f,0.f,0.f};
  v8f acc2 = {0.f,0.f,0.f,0.f,0.f,0.f,0.f,0.f};
#pragma unroll
  for (int k0 = 0; k0 < FF; k0 += 4) {
    v2f a = *(const v2f*)(xr + k0);
    a *= zm;
    const v2f a2 = a * a;
    acc  = __builtin_amdgcn_wmma_f32_16x16x4_f32(false, a,  false, ones2,
                                                 (short)0, acc,  false, false);
    acc2 = __builtin_amdgcn_wmma_f32_16x16x4_f32(false, a2, false, ones2,
                                                 (short)0, acc2, false, false);
  }

  // ---- Phase B: per-feature constants (features fixed per lane) ----
  const int f0 = lane * 4;
  const float invN = 1.0f / (float)NN;
  const float invF = 1.0f / (float)FF;
  const v4f zero4 = {0.f, 0.f, 0.f, 0.f};
  const v4f bsv = *(const v4f*)(bsum + f0);
  const v4f bqv = *(const v4f*)(bsq  + f0);
  const v4f bmu = bsv * invN;
  const v4f brs = rsqrt4(bqv * invN - bmu * bmu + EPSF);  // batch var: no clamp (matches ref)
  const v4f l0v = *(const v4f*)(lb  + f0);
  const v4f l1v = *(const v4f*)(lg  + f0);
  const v4f l2v = *(const v4f*)(la  + f0);
  const v4f l3v = *(const v4f*)(lnv + f0);
  const v4f mx  = max4(max4(l0v, l1v), max4(l2v, l3v));
  const v4f e0 = exp4(l0v - mx), e1 = exp4(l1v - mx);
  const v4f e2 = exp4(l2v - mx), e3 = exp4(l3v - mx);
  const v4f es = e0 + e1 + e2 + e3;
  const v4f w0 = e0 / es, w1 = e1 / es, w2 = e2 / es, w3 = e3 / es;
  const v4f gam = *(const v4f*)(gamma + f0);
  const v4f bet = *(const v4f*)(beta  + f0);

#pragma unroll
  for (int r = 0; r < 16; ++r) {
    const int row = tb + r;
    if (row >= NN) break;
    // row-sum for row r lives in lanes {0..15} (r<8) or {16..31} (r>=8), VGPR r&7
    const float rs1 = __shfl(acc [r & 7], (r < 8) ? 0 : 16, 32);
    const float rq1 = __shfl(acc2[r & 7], (r < 8) ? 0 : 16, 32);
    const float nmu = rs1 * invF;
    const float nrs = rsqrtf(rq1 * invF - nmu * nmu + EPSF);
    const int gid   = gidv[row];
    const float ic  = 1.0f / fmaxf(gcnt[gid], 1.0f);
    const float idg = 1.0f / fmaxf(deg[row], 1.0f);
    const v4f xv  = *(const v4f*)(x    + (size_t)row * FF + f0);
    const v4f gsv = *(const v4f*)(gsum + (size_t)gid * FF + f0);
    const v4f gqv = *(const v4f*)(gsq  + (size_t)gid * FF + f0);
    const v4f asv = *(const v4f*)(asum + (size_t)row * FF + f0);
    const v4f aqv = *(const v4f*)(asq  + (size_t)row * FF + f0);
    const v4f gmu = gsv * ic;
    const v4f grs = rsqrt4(max4(gqv * ic - gmu * gmu, zero4) + EPSF);
    const v4f amu = asv * idg;
    const v4f ars = rsqrt4(max4(aqv * idg - amu * amu, zero4) + EPSF);
    const v4f xb = (xv - bmu) * brs;
    const v4f xg = (xv - gmu) * grs;
    const v4f xa = (xv - amu) * ars;
    const v4f xn = (xv - nmu) * nrs;
    const v4f o  = gam * (w0 * xb + w1 * xg + w2 * xa + w3 * xn) + bet;
    *(v4f*)(out + (size_t)row * FF + f0) = o;
  }
}

extern "C" void kernel_launch(void* const* d_in, const int* in_sizes, int n_in,
                              void* d_out, int out_size, void* d_ws, size_t ws_size,
                              hipStream_t stream) {
  const float* x     = (const float*)d_in[0];
  const float* gamma = (const float*)d_in[1];
  const float* beta  = (const float*)d_in[2];
  const float* lb    = (const float*)d_in[3];
  const float* lg    = (const float*)d_in[4];
  const float* la    = (const float*)d_in[5];
  const float* lnv   = (const float*)d_in[6];
  const int*   gid   = (const int*)d_in[7];
  const int*   esrc  = (const int*)d_in[8];
  const int*   edst  = (const int*)d_in[9];
  float* out = (float*)d_out;
  float* ws  = (float*)d_ws;

  float* bsum = ws + OFF_BSUM;
  float* bsq  = ws + OFF_BSQ;
  float* gcnt = ws + OFF_GCNT;
  float* gsum = ws + OFF_GSUM;
  float* gsq  = ws + OFF_GSQ;
  float* deg  = ws + OFF_DEG;
  float* as   = ws + OFF_AS;
  float* as2  = ws + OFF_AS2;

  hipMemsetAsync(d_ws, 0, WS_FLOATS * sizeof(float), stream);

  k_stats<<<(NN + 63) / 64, 256, 0, stream>>>(x, gid, bsum, bsq, gcnt, gsum, gsq);
  k_edges<<<(EE + 31) / 32, 256, 0, stream>>>(x, esrc, edst, as, as2, deg);
  k_final<<<((NN / 16) + 7) / 8, 256, 0, stream>>>(
      x, gamma, beta, lb, lg, la, lnv, gid,
      bsum, bsq, gcnt, gsum, gsq, deg, as, as2, out);
  (void)in_sizes; (void)n_in; (void)out_size; (void)ws_size;
}